// DoublyStochasticButterfly_16466904613529
// MI455X (gfx1250) — compile-verified
//
#include <hip/hip_runtime.h>

#define WIDTHX 1024
#define HALFX  512
#define DEPTHX 20
#define BATCHX 32768

#if defined(__has_builtin)
#if __has_builtin(__builtin_amdgcn_global_load_async_to_lds_b128)
#define HAVE_ASYNC_LDS 1
#endif
#endif

typedef int v4i __attribute__((ext_vector_type(4)));
typedef __attribute__((address_space(1))) v4i* gptr4;   // global int4*
typedef __attribute__((address_space(3))) v4i* lptr4;   // LDS int4*

// ---- bit helpers -----------------------------------------------------------
__device__ __forceinline__ unsigned insert0(unsigned x, int b) {
    return ((x >> b) << (b + 1)) | (x & ((1u << b) - 1u));
}
__device__ __forceinline__ unsigned rotl10(unsigned q, int s) {
    s %= 10;
    if (s == 0) return q & 1023u;
    return ((q << s) | (q >> (10 - s))) & 1023u;
}

// Step d uses butterfly bit k = (9 - d%10).
// Physical placement of q bits in the (reg, lane) layout:
//   q9,q8,q7 -> reg bits 4,3,2   (q = ((r>>2)<<7) | (lane<<2) | (r&3))
//   q6..q2   -> lane bits 4..0
//   q1,q0    -> reg bits 1,0

// ---- prologue: pre-permute params into per-step load order -----------------
__global__ void dsb_build_w(const float* __restrict__ params,
                            float* __restrict__ Wg) {
    int idx = blockIdx.x * blockDim.x + threadIdx.x;   // 0 .. 10239
    if (idx >= DEPTHX * HALFX) return;
    int d = idx / HALFX;
    int i = idx % HALFX;
    int k = 9 - (d % 10);
    unsigned q;
    if (k >= 7 || k <= 1) {               // register-xor step
        int b = (k >= 7) ? (k - 5) : k;
        unsigned pi = (unsigned)(i >> 5), l = (unsigned)(i & 31);
        unsigned r0 = insert0(pi, b);     // reg with bit b == 0
        q = ((r0 >> 2) << 7) | (l << 2) | (r0 & 3);
    } else {                              // lane-xor step
        int b = k - 2;
        unsigned r = (unsigned)(i >> 4), cl = (unsigned)(i & 15);
        unsigned l = insert0(cl, b);      // lane with bit b == 0
        q = ((r >> 2) << 7) | (l << 2) | (r & 3);
    }
    unsigned j = rotl10(q, d + 1) >> 1;   // weight index for this pair
    Wg[idx] = params[j * DEPTHX + d];     // params is (HALF, DEPTH) row-major
}

// ---- main kernel: 256 threads (8 waves), 2 rows per wave, 16 rows/block ----
__global__ __launch_bounds__(256, 6) void dsb_main(const float* __restrict__ X,
                                                   const float* __restrict__ Wg,
                                                   float* __restrict__ Out) {
    __shared__ float Wl[DEPTHX * HALFX];              // 40 KB
    const int tid  = threadIdx.x;
    const int lane = tid & 31;
    const int wid  = tid >> 5;

    // stage pre-permuted weights into LDS: 2560 x b128 (async DMA, ASYNCcnt)
#if HAVE_ASYNC_LDS
    {
        unsigned long long gbase = (unsigned long long)(const void*)Wg;
        unsigned int lbase = (unsigned int)(unsigned long long)(const void*)Wl;
        #pragma unroll
        for (int t = 0; t < 10; ++t) {
            int idx4 = t * 256 + tid;
            __builtin_amdgcn_global_load_async_to_lds_b128(
                (gptr4)(gbase + (unsigned long long)idx4 * 16ull),
                (lptr4)(lbase + (unsigned int)idx4 * 16u),
                0, 0);
        }
#if defined(__has_builtin) && __has_builtin(__builtin_amdgcn_s_wait_asynccnt)
        __builtin_amdgcn_s_wait_asynccnt(0);
#else
        asm volatile("s_wait_asynccnt 0" ::: "memory");
#endif
    }
#else
    {
        const float4* Wg4 = (const float4*)Wg;
        float4* Wl4 = (float4*)Wl;
        #pragma unroll
        for (int t = 0; t < 10; ++t)
            Wl4[t * 256 + tid] = Wg4[t * 256 + tid];
    }
#endif
    __syncthreads();

    // compressed-lane index (lane with bit b removed), per lane-bit b
    int cl[5];
    #pragma unroll
    for (int b = 0; b < 5; ++b)
        cl[b] = (int)((((unsigned)lane >> (b + 1)) << b) |
                      ((unsigned)lane & ((1u << b) - 1u)));

    // load 2 rows, fully coalesced b128; z[rr][r] holds element
    // q = ((r>>2)<<7) | (lane<<2) | (r&3) of row rr
    float z[2][32];
    const int rowBase = blockIdx.x * 16 + wid * 2;
    const float4* Xv = (const float4*)X;
    #pragma unroll
    for (int rr = 0; rr < 2; ++rr) {
        const int base = (rowBase + rr) * 256;
        #pragma unroll
        for (int m = 0; m < 8; ++m) {
            float4 v = Xv[base + m * 32 + lane];
            z[rr][m * 4 + 0] = v.x;
            z[rr][m * 4 + 1] = v.y;
            z[rr][m * 4 + 2] = v.z;
            z[rr][m * 4 + 3] = v.w;
        }
    }

    // 20 butterfly steps, fully unrolled; z'[q] = z[q] + w*(z[q^2^k] - z[q])
    #pragma unroll
    for (int d = 0; d < 20; ++d) {
        const int k = 9 - (d % 10);
        if (k >= 7 || k <= 1) {
            // register-xor step on reg bit b
            const int b = (k >= 7) ? (k - 5) : k;
            #pragma unroll
            for (int pi = 0; pi < 16; ++pi) {
                const int r0 = (int)insert0((unsigned)pi, b);
                const int r1 = r0 | (1 << b);
                const float w = Wl[d * 512 + pi * 32 + lane];
                #pragma unroll
                for (int rr = 0; rr < 2; ++rr) {
                    const float diff = z[rr][r1] - z[rr][r0];
                    z[rr][r0] = fmaf(w, diff, z[rr][r0]);
                    z[rr][r1] = fmaf(-w, diff, z[rr][r1]);
                }
            }
        } else {
            // lane-xor step on lane bit b (cross-lane via ds_swizzle)
            const int b = k - 2;
            #pragma unroll
            for (int r = 0; r < 32; ++r) {
                const float w = Wl[d * 512 + r * 16 + cl[b]];
                #pragma unroll
                for (int rr = 0; rr < 2; ++rr) {
                    const float zp = __shfl_xor(z[rr][r], 1 << b, 32);
                    z[rr][r] = fmaf(w, zp - z[rr][r], z[rr][r]);
                }
            }
        }
    }

    // store in natural order (rotation is identity after 20 steps)
    float4* Ov = (float4*)Out;
    #pragma unroll
    for (int rr = 0; rr < 2; ++rr) {
        const int base = (rowBase + rr) * 256;
        #pragma unroll
        for (int m = 0; m < 8; ++m) {
            float4 v;
            v.x = z[rr][m * 4 + 0];
            v.y = z[rr][m * 4 + 1];
            v.z = z[rr][m * 4 + 2];
            v.w = z[rr][m * 4 + 3];
            Ov[base + m * 32 + lane] = v;
        }
    }
}

extern "C" void kernel_launch(void* const* d_in, const int* in_sizes, int n_in,
                              void* d_out, int out_size, void* d_ws, size_t ws_size,
                              hipStream_t stream) {
    const float* X      = (const float*)d_in[0];   // (32768, 1024) fp32
    const float* params = (const float*)d_in[1];   // (512, 20) fp32
    float* Out = (float*)d_out;
    float* Wg  = (float*)d_ws;                     // 40 KB scratch for permuted weights

    dsb_build_w<<<(DEPTHX * HALFX) / 256, 256, 0, stream>>>(params, Wg);
    dsb_main<<<BATCHX / 16, 256, 0, stream>>>(X, Wg, Out);
}